// MyelinModel_89773406421260
// MI455X (gfx1250) — compile-verified
//
#include <hip/hip_runtime.h>

// V_WMMA_F32_16X16X4_F32 (CDNA5 ISA layouts):
//   A 16x4 f32 : v2f/lane. lanes 0-15: M=lane, localK={0,1}; lanes 16-31: M=lane-16, localK={2,3}
//   B 4x16 f32 : v2f/lane. lanes 0-15: N=lane, localK={0,1}; lanes 16-31: N=lane-16, localK={2,3}
//   C/D 16x16  : v8f/lane. VGPR r: lanes 0-15 = (M=r, N=lane); lanes 16-31 = (M=8+r, N=lane-16)
//
// K-permutation trick: split the k=0..7 reduction as WMMA#0 -> global k {0,1,4,5},
// WMMA#1 -> global k {2,3,6,7} (A mask permuted to match). A lane (half h) then needs
// floats [4h .. 4h+3] of its row: ONE b128 load feeds both WMMAs.
typedef __attribute__((ext_vector_type(2))) float v2f;
typedef __attribute__((ext_vector_type(4))) float v4f;
typedef __attribute__((ext_vector_type(8))) float v8f;
typedef __attribute__((ext_vector_type(4))) int   v4i;

// pred[row] = 1 + 4 * first_argmax_j( sum_{k>=j} x[row][k] )   (see round-0 reduction:
// mask is constant tril, softmax is monotonic, pos identical across the 4 iterations).
// Computed as D = tril^T x^T so each lane holds one full output row -> lane-local argmax.
__global__ __launch_bounds__(256) void decode_pred_wmma(const float* __restrict__ x,
                                                        int* __restrict__ pred,
                                                        long nrows)
{
    const int  lane = (int)(threadIdx.x & 31u);
    const long wave = ((long)blockIdx.x * (long)blockDim.x + (long)threadIdx.x) >> 5;
    const int  half = lane >> 4;
    const int  n    = lane & 15;

    const long base = wave * 64;          // 64 rows = 8 matrices = 2 KB per wave
    if (base >= nrows) return;            // wave-uniform

    if (base + 64 <= nrows) {
        // ---------- fast path: full tile, no per-lane guards, EXEC all-ones ----------
        // A = tril^T with the permuted K map: entry = (j <= global_k) ? 1 : 0; j>=8 rows -> 0.
        const int j = n;
        v2f a0, a1;
        a0.x = (j <= 4 * half + 0) ? 1.0f : 0.0f;   // WMMA#0: global k = {0,1,4,5}
        a0.y = (j <= 4 * half + 1) ? 1.0f : 0.0f;
        a1.x = (j <= 4 * half + 2) ? 1.0f : 0.0f;   // WMMA#1: global k = {2,3,6,7}
        a1.y = (j <= 4 * half + 3) ? 1.0f : 0.0f;

        // Sub-tile s (s=0..3) covers rows base + 4*n + s; per lane: one b128 at
        // x + (base + 4n + s)*8 + 4*half  -> common base + immediate offsets 0,32,64,96 B.
        const float* p = x + base * 8 + (long)n * 32 + half * 4;
        v4f f0 = __builtin_nontemporal_load((const v4f*)(p + 0));
        v4f f1 = __builtin_nontemporal_load((const v4f*)(p + 8));
        v4f f2 = __builtin_nontemporal_load((const v4f*)(p + 16));
        v4f f3 = __builtin_nontemporal_load((const v4f*)(p + 24));

        v2f b00 = {f0.x, f0.y}, b01 = {f0.z, f0.w};
        v2f b10 = {f1.x, f1.y}, b11 = {f1.z, f1.w};
        v2f b20 = {f2.x, f2.y}, b21 = {f2.z, f2.w};
        v2f b30 = {f3.x, f3.y}, b31 = {f3.z, f3.w};

        v8f d0 = {}, d1 = {}, d2 = {}, d3 = {};
        d0 = __builtin_amdgcn_wmma_f32_16x16x4_f32(false, a0, false, b00, (short)0, d0, false, false);
        d1 = __builtin_amdgcn_wmma_f32_16x16x4_f32(false, a0, false, b10, (short)0, d1, false, false);
        d2 = __builtin_amdgcn_wmma_f32_16x16x4_f32(false, a0, false, b20, (short)0, d2, false, false);
        d3 = __builtin_amdgcn_wmma_f32_16x16x4_f32(false, a0, false, b30, (short)0, d3, false, false);
        d0 = __builtin_amdgcn_wmma_f32_16x16x4_f32(false, a1, false, b01, (short)0, d0, false, false);
        d1 = __builtin_amdgcn_wmma_f32_16x16x4_f32(false, a1, false, b11, (short)0, d1, false, false);
        d2 = __builtin_amdgcn_wmma_f32_16x16x4_f32(false, a1, false, b21, (short)0, d2, false, false);
        d3 = __builtin_amdgcn_wmma_f32_16x16x4_f32(false, a1, false, b31, (short)0, d3, false, false);

        // Lanes 0-15: d[r] = suffix sum starting at j=r of row (base + 4n + s).
        // First-max tie-break (strictly-greater update) matches jnp.argmax.
        auto amax = [](const v8f& d) -> int {
            float m = d[0]; int bi = 0;
#pragma unroll
            for (int r = 1; r < 8; ++r) { if (d[r] > m) { m = d[r]; bi = r; } }
            return 1 + 4 * bi;
        };
        v4i res = { amax(d0), amax(d1), amax(d2), amax(d3) };   // rows 4n+0..4n+3: contiguous

        if (half == 0)   // lanes 0-15 store 16B each -> one 256 B contiguous NT store
            __builtin_nontemporal_store(res, (v4i*)(pred + base + 4 * n));
    } else {
        // ---------- tail path (at most one wave): plain per-lane scalar version ----------
#pragma unroll
        for (int t = 0; t < 2; ++t) {
            const long r = base + t * 32 + lane;
            if (r < nrows) {
                const float* rp = x + r * 8;
                v4f lo = __builtin_nontemporal_load((const v4f*)rp);
                v4f hi = __builtin_nontemporal_load((const v4f*)(rp + 4));
                const float s7 = hi.w,      s6 = s7 + hi.z, s5 = s6 + hi.y, s4 = s5 + hi.x;
                const float s3 = s4 + lo.w, s2 = s3 + lo.z, s1 = s2 + lo.y, s0 = s1 + lo.x;
                const float sv[8] = {s0, s1, s2, s3, s4, s5, s6, s7};
                float m = sv[0]; int bi = 0;
#pragma unroll
                for (int q = 1; q < 8; ++q) { if (sv[q] > m) { m = sv[q]; bi = q; } }
                pred[r] = 1 + 4 * bi;
            }
        }
    }
}

extern "C" void kernel_launch(void* const* d_in, const int* in_sizes, int n_in,
                              void* d_out, int out_size, void* d_ws, size_t ws_size,
                              hipStream_t stream) {
    const float* x    = (const float*)d_in[0];
    int*         pred = (int*)d_out;              // reference output dtype is int32

    const long nrows = (long)in_sizes[0] / 8;     // (B,8,8) -> B*8 rows of 8 floats
    const long waves = (nrows + 63) / 64;         // 64 rows per wave
    const int  wavesPerBlock = 256 / 32;
    const long blocks = (waves + wavesPerBlock - 1) / wavesPerBlock;

    decode_pred_wmma<<<(unsigned)blocks, 256, 0, stream>>>(x, pred, nrows);
}